// SpatialMix_46755013984909
// MI455X (gfx1250) — compile-verified
//
#include <hip/hip_runtime.h>
#include <math.h>

#define B_   2
#define C_   256
#define N_   100000
#define H_   256
#define W_   256
#define HW_  65536
#define EPS_ 1e-5f

// ---------------------------------------------------------------------------
// CDNA5 async global->LDS copy (ASYNCcnt-tracked). Guarded so compile always
// succeeds; falls back to a plain float4 copy if the builtin is absent.
// ---------------------------------------------------------------------------
#if defined(__HIP_DEVICE_COMPILE__) && __has_builtin(__builtin_amdgcn_global_load_async_to_lds_b128)
#define USE_ASYNC_LDS 1
#else
#define USE_ASYNC_LDS 0
#endif

typedef int v4i_t __attribute__((ext_vector_type(4)));
typedef __attribute__((address_space(1))) v4i_t* global_v4i_p;
typedef __attribute__((address_space(3))) v4i_t* lds_v4i_p;

__device__ __forceinline__ void async_copy16(const void* g, void* l) {
#if USE_ASYNC_LDS
  __builtin_amdgcn_global_load_async_to_lds_b128((global_v4i_p)g, (lds_v4i_p)l, 0, 0);
#else
  *(float4*)l = *(const float4*)g;
#endif
}

__device__ __forceinline__ void async_wait_all() {
#if USE_ASYNC_LDS
  asm volatile("s_wait_asynccnt 0" ::: "memory");
#endif
}

__device__ __forceinline__ void atomicAddF32(float* p, float v) {
  __hip_atomic_fetch_add(p, v, __ATOMIC_RELAXED, __HIP_MEMORY_SCOPE_AGENT);
}

// ---------------------------------------------------------------------------
// Zero workspace region (grid accumulator + counts), float4 grid-stride.
// Default (RT) stores: we WANT the grid accumulator resident in the 192MB L2
// so the scatter atomics and conv reads hit L2.
// ---------------------------------------------------------------------------
__global__ __launch_bounds__(256) void k_zero(float4* __restrict__ p, long n4) {
  long i = (long)blockIdx.x * blockDim.x + threadIdx.x;
  const long stride = (long)gridDim.x * blockDim.x;
  const float4 z = make_float4(0.f, 0.f, 0.f, 0.f);
  for (; i < n4; i += stride) p[i] = z;
}

// ---------------------------------------------------------------------------
// LayerNorm over C=256 per token + scatter-add into grid[B,HW,C] (+counts).
// Block = 32 tokens. Phase 1: coalesced 128B global loads (non-temporal:
// tokens are streamed once here), LDS transpose. Phase 2: wave-per-token
// shfl reduction, contiguous 128B f32 atomics (L2-resident grid).
// ---------------------------------------------------------------------------
__global__ __launch_bounds__(256) void k_ln_scatter(
    const float* __restrict__ tokens, const int* __restrict__ fidx,
    const float* __restrict__ lnw, const float* __restrict__ lnb,
    float* __restrict__ grid, float* __restrict__ cnt)
{
  __shared__ float xs[32 * 257];     // 32 tokens x 256 ch, padded (stride 257)
  __shared__ float lnw_s[256];
  __shared__ float lnb_s[256];
  const int tid = threadIdx.x;
  const int w = tid >> 5, l = tid & 31;
  const int b = blockIdx.y;
  const long n0 = (long)blockIdx.x * 32;    // 3125*32 == 100000, no remainder

  lnw_s[tid] = lnw[tid];
  lnb_s[tid] = lnb[tid];

  const float* tb = tokens + (long)b * C_ * N_ + n0;
#pragma unroll 4
  for (int it = 0; it < 32; ++it) {
    const int c = it * 8 + w;                 // wave-uniform channel
    xs[l * 257 + c] = __builtin_nontemporal_load(&tb[(long)c * N_ + l]);
  }
  __syncthreads();

  for (int j = 0; j < 4; ++j) {
    const int t = w * 4 + j;                  // token within tile
    const long n = n0 + t;
    float x[8];
    float s1 = 0.f, s2 = 0.f;
#pragma unroll
    for (int k = 0; k < 8; ++k) {
      x[k] = xs[t * 257 + l + 32 * k];
      s1 += x[k];
      s2 += x[k] * x[k];
    }
#pragma unroll
    for (int off = 16; off > 0; off >>= 1) {
      s1 += __shfl_xor(s1, off, 32);
      s2 += __shfl_xor(s2, off, 32);
    }
    const float mu  = s1 * (1.f / 256.f);
    const float var = s2 * (1.f / 256.f) - mu * mu;
    const float rs  = rsqrtf(var + EPS_);
    const int idx = fidx[(long)b * N_ + n];
    float* gcell = grid + ((long)b * HW_ + idx) * C_;
#pragma unroll
    for (int k = 0; k < 8; ++k) {
      const int c = l + 32 * k;
      const float xn = (x[k] - mu) * rs * lnw_s[c] + lnb_s[c];
      atomicAddF32(gcell + c, xn);            // lanes -> 128B contiguous
    }
    if (l == 0) atomicAddF32(cnt + b * HW_ + idx, 1.0f);
  }
}

// ---------------------------------------------------------------------------
// Depthwise 3x3 conv over grid[B,HW,C] (SAME pad). Output tile 4x8 cells,
// input tile 6x10 cells (60KB) staged via async global->LDS b128.
// Optional: prescale by 1/max(cnt,1) (conv1), ReLU, per-channel scale (conv2).
// One thread == one channel; all global traffic is 128B-coalesced across c.
// ---------------------------------------------------------------------------
__global__ __launch_bounds__(256) void k_dwconv(
    const float* __restrict__ in, const float* __restrict__ cntp,
    const float* __restrict__ wts, const float* __restrict__ bias,
    const float* __restrict__ scalep, int do_relu,
    float* __restrict__ out)
{
  __shared__ float tile_s[60 * 256];   // 60 cells x 1KB = 60KB
  __shared__ float rc_s[60];
  const int tid = threadIdx.x;
  const int w0 = blockIdx.x * 8;
  const int h0 = blockIdx.y * 4;
  const int b  = blockIdx.z;
  const long base = (long)b * HW_;

  // Stage 60 cells (3840 x 16B chunks, 15 per thread), zero-fill OOB halo.
  for (int q = tid; q < 3840; q += 256) {
    const int cell = q >> 6;              // 64 chunks per cell row
    const int co   = (q & 63) << 2;       // float offset within cell row
    const int r = cell / 10, ccol = cell % 10;
    const int ch = h0 - 1 + r, cw = w0 - 1 + ccol;
    float* ld = &tile_s[cell * 256 + co];
    if (ch >= 0 && ch < H_ && cw >= 0 && cw < W_) {
      const float* gs = in + (base + (long)ch * W_ + cw) * C_ + co;
      async_copy16(gs, ld);
    } else {
      ld[0] = 0.f; ld[1] = 0.f; ld[2] = 0.f; ld[3] = 0.f;
    }
  }
  if (cntp != nullptr && tid < 60) {
    const int r = tid / 10, ccol = tid % 10;
    const int ch = h0 - 1 + r, cw = w0 - 1 + ccol;
    float rc = 1.0f;
    if (ch >= 0 && ch < H_ && cw >= 0 && cw < W_)
      rc = 1.0f / fmaxf(cntp[base + (long)ch * W_ + cw], 1.0f);
    rc_s[tid] = rc;
  }
  async_wait_all();
  __syncthreads();

  if (cntp != nullptr) {                  // scatter-mean division (conv1 only)
    for (int cell = 0; cell < 60; ++cell)
      tile_s[cell * 256 + tid] *= rc_s[cell];
    __syncthreads();
  }

  float wk[9];
#pragma unroll
  for (int k = 0; k < 9; ++k) wk[k] = wts[tid * 9 + k];
  const float bb = bias[tid];
  const float sc = (scalep != nullptr) ? scalep[tid] : 1.0f;

  for (int oy = 0; oy < 4; ++oy) {
    for (int ox = 0; ox < 8; ++ox) {
      float acc = bb;
#pragma unroll
      for (int kh = 0; kh < 3; ++kh)
#pragma unroll
        for (int kw = 0; kw < 3; ++kw)
          acc += wk[kh * 3 + kw] * tile_s[((oy + kh) * 10 + (ox + kw)) * 256 + tid];
      if (do_relu) acc = fmaxf(acc, 0.0f);
      acc *= sc;
      out[(base + (long)(h0 + oy) * W_ + (w0 + ox)) * C_ + tid] = acc;
    }
  }
}

// ---------------------------------------------------------------------------
// Gather final grid rows per token (contiguous 1KB reads, L2-resident),
// LDS transpose, residual add, coalesced [B,C,N] writes.
// Tokens re-read and output written non-temporally (stream-once data) so the
// final grid keeps the L2 for its random per-token row gathers.
// ---------------------------------------------------------------------------
__global__ __launch_bounds__(256) void k_gather_add(
    const float* __restrict__ tokens, const int* __restrict__ iidx,
    const float* __restrict__ fin, float* __restrict__ outp)
{
  __shared__ float xs[32 * 257];
  const int tid = threadIdx.x;
  const int w = tid >> 5, l = tid & 31;
  const int b = blockIdx.y;
  const long n0 = (long)blockIdx.x * 32;

  for (int j = 0; j < 4; ++j) {
    const int t = w * 4 + j;
    const int idx = iidx[(long)b * N_ + n0 + t];
    const float* src = fin + ((long)b * HW_ + idx) * C_;
#pragma unroll
    for (int k = 0; k < 8; ++k)
      xs[t * 257 + l + 32 * k] = src[l + 32 * k];   // 128B contiguous, RT (L2)
  }
  __syncthreads();

#pragma unroll 4
  for (int it = 0; it < 32; ++it) {
    const int c = it * 8 + w;
    const long o = ((long)b * C_ + c) * N_ + n0 + l;
    const float tk = __builtin_nontemporal_load(&tokens[o]);
    __builtin_nontemporal_store(tk + xs[l * 257 + c], &outp[o]);
  }
}

// ---------------------------------------------------------------------------
extern "C" void kernel_launch(void* const* d_in, const int* in_sizes, int n_in,
                              void* d_out, int out_size, void* d_ws, size_t ws_size,
                              hipStream_t stream)
{
  const float* tokens = (const float*)d_in[0];
  const int*   fidx   = (const int*)d_in[1];
  const int*   iidx   = (const int*)d_in[2];
  const float* lnw    = (const float*)d_in[3];
  const float* lnb    = (const float*)d_in[4];
  const float* c1w    = (const float*)d_in[5];
  const float* c1b    = (const float*)d_in[6];
  const float* c2w    = (const float*)d_in[7];
  const float* c2b    = (const float*)d_in[8];
  const float* scw    = (const float*)d_in[9];
  float* outp = (float*)d_out;

  char* ws = (char*)d_ws;
  const size_t gridBytes = (size_t)B_ * HW_ * C_ * sizeof(float); // 134,217,728
  const size_t cntBytes  = (size_t)B_ * HW_ * sizeof(float);      //     524,288
  const size_t needed    = 2 * gridBytes + cntBytes;              // ~269 MB
  if (ws_size < needed) return;

  float* buf0 = (float*)ws;                            // grid sum / final grid
  float* cnt  = (float*)(ws + gridBytes);              // scatter counts
  float* buf1 = (float*)(ws + gridBytes + cntBytes);   // conv1 output (mid)

  // 1) zero grid accumulator + counts (contiguous region)
  const long n4 = (long)((gridBytes + cntBytes) / 16);
  k_zero<<<1024, 256, 0, stream>>>((float4*)ws, n4);

  // 2) LayerNorm + scatter-add
  dim3 tg(N_ / 32, B_, 1);   // 3125 x 2
  k_ln_scatter<<<tg, 256, 0, stream>>>(tokens, fidx, lnw, lnb, buf0, cnt);

  // 3) conv1: divide-by-count fused, ReLU
  dim3 cg(W_ / 8, H_ / 4, B_);  // 32 x 64 x 2
  k_dwconv<<<cg, 256, 0, stream>>>(buf0, cnt, c1w, c1b, nullptr, 1, buf1);

  // 4) conv2 + per-channel scale (writes back into buf0)
  k_dwconv<<<cg, 256, 0, stream>>>(buf1, nullptr, c2w, c2b, scw, 0, buf0);

  // 5) gather + residual
  k_gather_add<<<tg, 256, 0, stream>>>(tokens, iidx, buf0, outp);
}